// gliam_62826781606166
// MI455X (gfx1250) — compile-verified
//
#include <hip/hip_runtime.h>

// Solve 80 independent Kronecker-structured 800x800 systems p x = v with
// p = kron(I_20, S) + kron(M_b, Ng), via matrix-free Richardson iteration:
//   X <- X + 0.5*(V - (X S^T + (M X) Ng^T))
// All GEMM work on V_WMMA_F32_16X16X4_F32 (fp32 matrix pipe; exact precision
// parity with the fp32 reference). One workgroup per batch, state in LDS.
//
// v2: (a) hoist the 32 loop-invariant operand fragments (M rows, S^T/Ng^T
// columns) into VGPRs -- __syncthreads() fences LDS so the compiler can't do
// this itself; halves per-iteration DS traffic. (b) double-buffer X to drop
// from 3 barriers/iteration to 2.

#define GS      20
#define PARAM   40
#define MP      32          // GS padded to multiple of 16
#define NP      48          // PARAM padded to multiple of 16
#define RICH_ITERS 48       // contraction <= ~0.46 -> far below fp32 eps
#define NTHREADS 192        // 6 wave32 waves: one 16x16 tile each (2x3 tiles)

typedef float v2f __attribute__((ext_vector_type(2)));
typedef float v8f __attribute__((ext_vector_type(8)));

// A fragment (16x4 f32): lanes 0-15 rows M=0..15 with K={0,1}; lanes 16-31 K={2,3}.
__device__ __forceinline__ v2f load_a_frag(const float* L, int ld, int row0, int k0, int lane) {
    const int r = row0 + (lane & 15);
    const int k = k0 + ((lane >> 4) << 1);
    v2f a;
    a.x = L[r * ld + k];
    a.y = L[r * ld + k + 1];
    return a;
}

// B fragment (4x16 f32): lanes 0-15 cols N=0..15 with K={0,1}; lanes 16-31 K={2,3}.
__device__ __forceinline__ v2f load_b_frag(const float* L, int ld, int k0, int col0, int lane) {
    const int c = col0 + (lane & 15);
    const int k = k0 + ((lane >> 4) << 1);
    v2f b;
    b.x = L[k * ld + c];
    b.y = L[(k + 1) * ld + c];
    return b;
}

__global__ __launch_bounds__(NTHREADS)
void kron_richardson_solve(const float* __restrict__ mat,
                           const float* __restrict__ val,
                           const float* __restrict__ selfintact,
                           const float* __restrict__ neigintact,
                           float* __restrict__ out)
{
    __shared__ float sM [MP * MP];   // mat[b,a] zero-padded 32x32
    __shared__ float sST[NP * NP];   // selfintact^T zero-padded 48x48
    __shared__ float sNT[NP * NP];   // neigintact^T zero-padded 48x48
    __shared__ float sX0[MP * NP];   // iterate, double-buffered 32x48
    __shared__ float sX1[MP * NP];
    __shared__ float sV [MP * NP];   // RHS
    __shared__ float sY [MP * NP];   // Y = M X

    const int tid  = threadIdx.x;
    const int lane = tid & 31;
    const int wave = tid >> 5;              // 0..5
    const int bz   = blockIdx.x;            // batch (B*A flattened)

    // ---- zero-init padded LDS ----
    for (int i = tid; i < MP * MP; i += NTHREADS) sM[i] = 0.0f;
    for (int i = tid; i < NP * NP; i += NTHREADS) { sST[i] = 0.0f; sNT[i] = 0.0f; }
    for (int i = tid; i < MP * NP; i += NTHREADS) {
        sX0[i] = 0.0f; sX1[i] = 0.0f; sV[i] = 0.0f; sY[i] = 0.0f;
    }
    __syncthreads();

    // ---- stage inputs into LDS ----
    const float* Mg = mat + (size_t)bz * (GS * GS);
    for (int i = tid; i < GS * GS; i += NTHREADS)
        sM[(i / GS) * MP + (i % GS)] = Mg[i];

    for (int i = tid; i < PARAM * PARAM; i += NTHREADS) {
        const int r = i / PARAM, c = i % PARAM;
        // (X S^T)[g,p] = sum_q S[p,q] X[g,q]  -> B-matrix is S^T
        sST[c * NP + r] = selfintact[i];
        sNT[c * NP + r] = neigintact[i];
    }

    const float* Vg = val + (size_t)bz * (GS * PARAM);
    for (int i = tid; i < GS * PARAM; i += NTHREADS) {
        const int r = i / PARAM, c = i % PARAM;
        const float v = Vg[i];
        sV[r * NP + c] = v;
        sX0[r * NP + c] = 0.5f * v;   // X0 = V/2 (p ~ 2I)
    }
    __syncthreads();

    // ---- per-wave 16x16 output tile of the 32x48 state ----
    const int row0 = (wave / 3) * 16;
    const int col0 = (wave % 3) * 16;
    const int dr   = (lane >> 4) << 3;      // D rows: VGPR i -> row i (+8 upper lanes)
    const int dc   = lane & 15;

    // ---- hoist loop-invariant fragments into VGPRs ----
    v2f aM[8];                              // M rows (A-side), K = 0..31
#pragma unroll
    for (int kk = 0; kk < 8; ++kk)
        aM[kk] = load_a_frag(sM, MP, row0, 4 * kk, lane);

    v2f bST[12], bNT[12];                   // S^T / Ng^T columns (B-side), K = 0..47
#pragma unroll
    for (int kk = 0; kk < 12; ++kk) {
        bST[kk] = load_b_frag(sST, NP, 4 * kk, col0, lane);
        bNT[kk] = load_b_frag(sNT, NP, 4 * kk, col0, lane);
    }

    float* sXc = sX0;                       // current iterate
    float* sXn = sX1;                       // next iterate

    for (int it = 0; it < RICH_ITERS; ++it) {
        // Phase 1: Y = M * X_cur   (K over 32)
        v8f acc = {};
#pragma unroll
        for (int kk = 0; kk < 8; ++kk) {
            v2f b = load_b_frag(sXc, NP, 4 * kk, col0, lane);
            acc = __builtin_amdgcn_wmma_f32_16x16x4_f32(
                false, aM[kk], false, b, (short)0, acc, false, false);
        }
#pragma unroll
        for (int i = 0; i < 8; ++i)
            sY[(row0 + dr + i) * NP + (col0 + dc)] = acc[i];
        __syncthreads();   // Y published

        // Phase 2: Z = X_cur * S^T + Y * Ng^T   (K over 48 each)
        v8f z = {};
#pragma unroll
        for (int kk = 0; kk < 12; ++kk) {
            v2f a = load_a_frag(sXc, NP, row0, 4 * kk, lane);
            z = __builtin_amdgcn_wmma_f32_16x16x4_f32(
                false, a, false, bST[kk], (short)0, z, false, false);
        }
#pragma unroll
        for (int kk = 0; kk < 12; ++kk) {
            v2f a = load_a_frag(sY, NP, row0, 4 * kk, lane);
            z = __builtin_amdgcn_wmma_f32_16x16x4_f32(
                false, a, false, bNT[kk], (short)0, z, false, false);
        }

        // Richardson update into the other buffer (no read/write hazard on X_cur)
#pragma unroll
        for (int i = 0; i < 8; ++i) {
            const int idx = (row0 + dr + i) * NP + (col0 + dc);
            sXn[idx] = sXc[idx] + 0.5f * (sV[idx] - z[i]);
        }
        __syncthreads();   // X_next published (also fences this iter's sY reads)

        float* t = sXc; sXc = sXn; sXn = t;
    }

    // ---- advrelu == clip(x, -1, 1), then write out ----
    float* og = out + (size_t)bz * (GS * PARAM);
    for (int i = tid; i < GS * PARAM; i += NTHREADS) {
        const int r = i / PARAM, c = i % PARAM;
        float x = sXc[r * NP + c];
        x = fminf(1.0f, fmaxf(-1.0f, x));
        og[i] = x;
    }
}

extern "C" void kernel_launch(void* const* d_in, const int* in_sizes, int n_in,
                              void* d_out, int out_size, void* d_ws, size_t ws_size,
                              hipStream_t stream) {
    (void)n_in; (void)out_size; (void)d_ws; (void)ws_size;
    const float* mat = (const float*)d_in[0];   // (8,10,20,20)
    const float* val = (const float*)d_in[1];   // (8,10,20,40)
    const float* S   = (const float*)d_in[2];   // (40,40)
    const float* Ng  = (const float*)d_in[3];   // (40,40)
    float* out = (float*)d_out;                 // (80,20,40) flat

    const int nbatch = in_sizes[0] / (GS * GS); // 80
    kron_richardson_solve<<<nbatch, NTHREADS, 0, stream>>>(mat, val, S, Ng, out);
}